// GraphAutoencoder_5016521802203
// MI455X (gfx1250) — compile-verified
//
#include <hip/hip_runtime.h>
#include <hip/hip_bf16.h>

typedef float v2f __attribute__((ext_vector_type(2)));
typedef float v8f __attribute__((ext_vector_type(8)));

#define BN_EPS 1e-5f

// ---------------------------------------------------------------------------
// Utility kernels
// ---------------------------------------------------------------------------
__global__ void zero_f32(float* p, int n) {
    int i = blockIdx.x * blockDim.x + threadIdx.x;
    if (i < n) p[i] = 0.0f;
}

__global__ void zero_i32(int* p, int n) {
    int i = blockIdx.x * blockDim.x + threadIdx.x;
    if (i < n) p[i] = 0;
}

// deg[dst] += 1 for every edge (int atomics; exact)
__global__ void deg_count(const int* __restrict__ dst, int* __restrict__ deg, int E) {
    int e = blockIdx.x * blockDim.x + threadIdx.x;
    if (e < E) atomicAdd(&deg[dst[e]], 1);
}

// dinv = rsqrt(deg + 1)   (self-loop added)
__global__ void dinv_kernel(const int* __restrict__ deg, float* __restrict__ dinv, int n) {
    int i = blockIdx.x * blockDim.x + threadIdx.x;
    if (i < n) dinv[i] = rsqrtf((float)deg[i] + 1.0f);
}

// ---------------------------------------------------------------------------
// WMMA f32 GEMM:  C[N x F] = act(A)[N x K] @ W[K x F]
// One wave owns a 16-row strip and ALL F columns: A fragments are loaded once
// and reused across NT = ceil(F/16) accumulators (single-pass A traffic).
// SCALED: A element -> fmax(v,0)*scale[k] + shift[k]   (fused ReLU+BatchNorm)
// GCN:    out = raw t,  agg = t * dinv[row]^2 + bias   (GCNConv self-loop init)
// else:   out = t + bias
// Rows must be a multiple of 16 (200000 = 12500*16). Columns guarded (F=8).
// ---------------------------------------------------------------------------
template <int K, int F, bool GCN, bool SCALED>
__global__ void wmma_gemm_t(const float* __restrict__ A, const float* __restrict__ W,
                            const float* __restrict__ bias,
                            float* __restrict__ out, float* __restrict__ agg,
                            const float* __restrict__ dinv,
                            const float* __restrict__ scale,
                            const float* __restrict__ shift,
                            int mTiles) {
    constexpr int NT = (F + 15) / 16;
    int wave = (int)((blockIdx.x * blockDim.x + threadIdx.x) >> 5);
    if (wave >= mTiles) return;                   // wave-uniform exit
    int lane  = threadIdx.x & 31;
    int l15   = lane & 15;
    int kHalf = (lane >> 4) * 2;                  // 0 or 2
    int row   = wave * 16 + l15;                  // A-fragment row for this lane

    v8f acc[NT] = {};
    const float* Ap = A + (size_t)row * K + kHalf;
    for (int k0 = 0; k0 < K; k0 += 4) {
        int k = k0 + kHalf;
        float a0 = Ap[k0];
        float a1 = Ap[k0 + 1];
        if constexpr (SCALED) {
            a0 = fmaxf(a0, 0.0f) * scale[k] + shift[k];
            a1 = fmaxf(a1, 0.0f) * scale[k + 1] + shift[k + 1];
        }
        v2f a;
        a.x = a0;
        a.y = a1;
#pragma unroll
        for (int nt = 0; nt < NT; ++nt) {
            int col = nt * 16 + l15;
            bool colOK = (F % 16 == 0) || (col < F);
            int colC = colOK ? col : (F - 1);     // clamped in-bounds address
            float b0 = W[(size_t)k * F + colC];
            float b1 = W[(size_t)(k + 1) * F + colC];
            v2f b;
            b.x = colOK ? b0 : 0.0f;
            b.y = colOK ? b1 : 0.0f;
            acc[nt] = __builtin_amdgcn_wmma_f32_16x16x4_f32(
                /*neg_a=*/false, a, /*neg_b=*/false, b,
                /*c_mod=*/(short)0, acc[nt], /*reuse_a=*/false, /*reuse_b=*/false);
        }
    }

    int rbase = wave * 16 + (lane >> 4) * 8;      // C/D layout: M = vgpr + 8*(lane>=16)
    float dv2[8];
    if constexpr (GCN) {
#pragma unroll
        for (int i = 0; i < 8; ++i) {
            float d = dinv[rbase + i];
            dv2[i] = d * d;
        }
    }
#pragma unroll
    for (int nt = 0; nt < NT; ++nt) {
        int col = nt * 16 + l15;
        if ((F % 16 != 0) && col >= F) continue;  // after WMMA: divergence is safe
        float bv = bias[col];
        if constexpr (GCN) {
#pragma unroll
            for (int i = 0; i < 8; ++i) {
                int r = rbase + i;
                float t = acc[nt][i];
                out[(size_t)r * F + col] = t;                 // raw t (edge gather src)
                agg[(size_t)r * F + col] = t * dv2[i] + bv;   // self-loop + bias init
            }
        } else {
#pragma unroll
            for (int i = 0; i < 8; ++i) {
                int r = rbase + i;
                out[(size_t)r * F + col] = acc[nt][i] + bv;
            }
        }
    }
}

// ---------------------------------------------------------------------------
// Edge scatter: agg[dst,f] += t[src,f] * dinv[src]*dinv[dst]
// One thread per (edge, float4 chunk): consecutive threads read consecutive
// chunks of the same src row (256B coalesced); atomics land in L2.
// ---------------------------------------------------------------------------
__global__ void edge_scatter(const int* __restrict__ src, const int* __restrict__ dst,
                             const float* __restrict__ dinv, const float* __restrict__ t,
                             float* __restrict__ agg, int E, int F, int total) {
    int tid = blockIdx.x * blockDim.x + threadIdx.x;
    if (tid >= total) return;
    int chunks = F >> 2;
    int e = tid / chunks;
    int c = (tid - e * chunks) << 2;
    int s = src[e];
    int d = dst[e];
    float w = dinv[s] * dinv[d];
    const float4 tv = *reinterpret_cast<const float4*>(t + (size_t)s * F + c);
    float* ap = agg + (size_t)d * F + c;
    atomicAdd(ap + 0, tv.x * w);
    atomicAdd(ap + 1, tv.y * w);
    atomicAdd(ap + 2, tv.z * w);
    atomicAdd(ap + 3, tv.w * w);
}

// ---------------------------------------------------------------------------
// Read-only stats of relu(a): per-feature sum / sum-of-squares.
// F must divide 256 and be a power of two (16/32/64).
// ---------------------------------------------------------------------------
__global__ void relu_stats(const float* __restrict__ a, float* __restrict__ gsum,
                           float* __restrict__ gsumsq, int Nrows, int F, int rowsPerBlock) {
    __shared__ float s[256];
    __shared__ float ss[256];
    int t = threadIdx.x;
    int f = t & (F - 1);
    int rlane = t / F;
    int perIter = 256 / F;
    int r0 = blockIdx.x * rowsPerBlock;
    int r1 = r0 + rowsPerBlock;
    if (r1 > Nrows) r1 = Nrows;
    float ls = 0.0f, lss = 0.0f;
    for (int r = r0 + rlane; r < r1; r += perIter) {
        float v = a[(size_t)r * F + f];
        v = fmaxf(v, 0.0f);
        ls += v;
        lss += v * v;
    }
    s[t] = ls;
    ss[t] = lss;
    __syncthreads();
    for (int off = 128; off >= F; off >>= 1) {
        if (t < off) { s[t] += s[t + off]; ss[t] += ss[t + off]; }
        __syncthreads();
    }
    if (t < F) {
        atomicAdd(&gsum[f], s[t]);
        atomicAdd(&gsumsq[f], ss[t]);
    }
}

// ---------------------------------------------------------------------------
// Per-feature BN coefficients:  normalized(relu(v)) = relu(v)*scale + shift
//   scale = g * rsqrt(var+eps),  shift = bb - mean*scale   (biased batch stats)
// ---------------------------------------------------------------------------
__global__ void bn_coeffs(const float* __restrict__ gsum, const float* __restrict__ gsumsq,
                          const float* __restrict__ g, const float* __restrict__ bb,
                          float* __restrict__ scale, float* __restrict__ shift,
                          int Nrows, int F) {
    int f = threadIdx.x;
    if (f >= F) return;
    float invN = 1.0f / (float)Nrows;
    float mean = gsum[f] * invN;
    float var  = gsumsq[f] * invN - mean * mean;
    float sc   = rsqrtf(var + BN_EPS) * g[f];
    scale[f] = sc;
    shift[f] = bb[f] - mean * sc;
}

// ---------------------------------------------------------------------------
// Host-side orchestration
// ---------------------------------------------------------------------------
static inline int ceil_div(long long a, long long b) { return (int)((a + b - 1) / b); }

template <int K, int F, bool GCN, bool SCALED>
static void launch_gemm(const float* A, const float* W, const float* bias, float* out,
                        float* agg, const float* dinv, const float* scale,
                        const float* shift, int Nrows, hipStream_t stream) {
    int mTiles = Nrows / 16;
    int blocks = ceil_div((long long)mTiles * 32, 256);
    wmma_gemm_t<K, F, GCN, SCALED><<<blocks, 256, 0, stream>>>(
        A, W, bias, out, agg, dinv, scale, shift, mTiles);
}

static void launch_stats(const float* buf, float* gsum, float* gsumsq, const float* g,
                         const float* bb, float* scale, float* shift, int Nrows, int F,
                         hipStream_t stream) {
    zero_f32<<<1, 256, 0, stream>>>(gsum, 256);   // gsum (128) + gsumsq (128), contiguous
    int rowsPerBlock = 1024;
    relu_stats<<<ceil_div(Nrows, rowsPerBlock), 256, 0, stream>>>(buf, gsum, gsumsq, Nrows,
                                                                  F, rowsPerBlock);
    bn_coeffs<<<1, 128, 0, stream>>>(gsum, gsumsq, g, bb, scale, shift, Nrows, F);
}

extern "C" void kernel_launch(void* const* d_in, const int* in_sizes, int n_in,
                              void* d_out, int out_size, void* d_ws, size_t ws_size,
                              hipStream_t stream) {
    const int IN = 128;
    const int N = in_sizes[0] / IN;        // 200000
    const int E = in_sizes[1] / 2;         // 3200000

    // ---- inputs (setup_inputs dict order, tuples/lists flattened in order) ----
    const float* x    = (const float*)d_in[0];
    const int*   ei   = (const int*)d_in[1];
    const int*   src  = ei;
    const int*   dst  = ei + E;
    const float* eW0  = (const float*)d_in[2];   const float* eb0 = (const float*)d_in[3];
    const float* eW1  = (const float*)d_in[4];   const float* eb1 = (const float*)d_in[5];
    const float* eW2  = (const float*)d_in[6];   const float* eb2 = (const float*)d_in[7];
    const float* eg0  = (const float*)d_in[8];   const float* ebb0 = (const float*)d_in[9];
    const float* eg1  = (const float*)d_in[10];  const float* ebb1 = (const float*)d_in[11];
    const float* eg2  = (const float*)d_in[12];  const float* ebb2 = (const float*)d_in[13];
    const float* Wl   = (const float*)d_in[14];  const float* bl  = (const float*)d_in[15];
    const float* dW0  = (const float*)d_in[16];  const float* db0 = (const float*)d_in[17];
    const float* dW1  = (const float*)d_in[18];  const float* db1 = (const float*)d_in[19];
    const float* dW2  = (const float*)d_in[20];  const float* db2 = (const float*)d_in[21];
    const float* dg0  = (const float*)d_in[22];  const float* dbb0 = (const float*)d_in[23];
    const float* dg1  = (const float*)d_in[24];  const float* dbb1 = (const float*)d_in[25];
    const float* dg2  = (const float*)d_in[26];  const float* dbb2 = (const float*)d_in[27];
    const float* Wo   = (const float*)d_in[28];  const float* bo  = (const float*)d_in[29];
    float* outp = (float*)d_out;

    // ---- workspace layout (all offsets 16B-aligned; N*4 = 800000 B, /16 ok) ----
    char* ws = (char*)d_ws;
    int*   degi   = (int*)ws;                    ws += (size_t)N * sizeof(int);
    float* dinv   = (float*)ws;                  ws += (size_t)N * sizeof(float);
    float* gsum   = (float*)ws;                  /* 128 */
    float* gsumsq = gsum + 128;                  ws += 256 * sizeof(float);
    float* bscale = (float*)ws;                  /* 128 */
    float* bshift = bscale + 128;                ws += 256 * sizeof(float);
    float* bufT   = (float*)ws;                  ws += (size_t)N * 64 * sizeof(float);
    float* bufA   = (float*)ws;                  ws += (size_t)N * 64 * sizeof(float);
    float* bufB   = (float*)ws;                  ws += (size_t)N * 64 * sizeof(float);
    (void)ws_size; (void)n_in; (void)out_size;

    // ---- degree / symmetric norm ----
    zero_i32<<<ceil_div(N, 256), 256, 0, stream>>>(degi, N);
    deg_count<<<ceil_div(E, 256), 256, 0, stream>>>(dst, degi, E);
    dinv_kernel<<<ceil_div(N, 256), 256, 0, stream>>>(degi, dinv, N);

    // ---- encoder layer 0: x[N,128] -> 64 (raw input, no fused act) ----
    launch_gemm<128, 64, true, false>(x, eW0, eb0, bufT, bufB, dinv, nullptr, nullptr, N, stream);
    {
        int total = E * (64 / 4);
        edge_scatter<<<ceil_div(total, 256), 256, 0, stream>>>(src, dst, dinv, bufT, bufB, E, 64, total);
    }
    launch_stats(bufB, gsum, gsumsq, eg0, ebb0, bscale, bshift, N, 64, stream);

    // ---- encoder layer 1: 64 -> 32 (fused ReLU+BN on A-load) ----
    launch_gemm<64, 32, true, true>(bufB, eW1, eb1, bufT, bufA, dinv, bscale, bshift, N, stream);
    {
        int total = E * (32 / 4);
        edge_scatter<<<ceil_div(total, 256), 256, 0, stream>>>(src, dst, dinv, bufT, bufA, E, 32, total);
    }
    launch_stats(bufA, gsum, gsumsq, eg1, ebb1, bscale, bshift, N, 32, stream);

    // ---- encoder layer 2: 32 -> 16 ----
    launch_gemm<32, 16, true, true>(bufA, eW2, eb2, bufT, bufB, dinv, bscale, bshift, N, stream);
    {
        int total = E * (16 / 4);
        edge_scatter<<<ceil_div(total, 256), 256, 0, stream>>>(src, dst, dinv, bufT, bufB, E, 16, total);
    }
    launch_stats(bufB, gsum, gsumsq, eg2, ebb2, bscale, bshift, N, 16, stream);

    // ---- latent: 16 -> 8 (fused ReLU+BN in, plain dense out) ----
    launch_gemm<16, 8, false, true>(bufB, Wl, bl, bufA, nullptr, nullptr, bscale, bshift, N, stream);

    // ---- decoder 0: 8 -> 16 (latent is raw, no fused act) ----
    launch_gemm<8, 16, false, false>(bufA, dW0, db0, bufT, nullptr, nullptr, nullptr, nullptr, N, stream);
    launch_stats(bufT, gsum, gsumsq, dg0, dbb0, bscale, bshift, N, 16, stream);

    // ---- decoder 1: 16 -> 32 ----
    launch_gemm<16, 32, false, true>(bufT, dW1, db1, bufA, nullptr, nullptr, bscale, bshift, N, stream);
    launch_stats(bufA, gsum, gsumsq, dg1, dbb1, bscale, bshift, N, 32, stream);

    // ---- decoder 2: 32 -> 64 ----
    launch_gemm<32, 64, false, true>(bufA, dW2, db2, bufT, nullptr, nullptr, bscale, bshift, N, stream);
    launch_stats(bufT, gsum, gsumsq, dg2, dbb2, bscale, bshift, N, 64, stream);

    // ---- output: 64 -> 128, fused ReLU+BN in, straight to d_out ----
    launch_gemm<64, 128, false, true>(bufT, Wo, bo, outp, nullptr, nullptr, bscale, bshift, N, stream);
}